// PointNetFeaturePropagation_59717225284092
// MI455X (gfx1250) — compile-verified
//
#include <hip/hip_runtime.h>

typedef float v2f __attribute__((ext_vector_type(2)));
typedef float v8f __attribute__((ext_vector_type(8)));
typedef int   v4i __attribute__((ext_vector_type(4)));

#define B_    16
#define N_    4096
#define S_    1024
#define D1_   128
#define D2_   256
#define CIN   384      // D1 + D2
#define CMID  256
#define COUT  128
#define M_TOT (B_ * N_)   // 65536
#define BN_EPS 1e-5f

// CDNA5 async global->LDS copy (tracked by ASYNCcnt), direct LDS write.
// Builtin signature (from clang diagnostic): (v4i AS1*, v4i AS3*, imm, imm)
#define ASYNC_LOAD_B128(gptr, lptr)                                          \
  __builtin_amdgcn_global_load_async_to_lds_b128(                            \
      (__attribute__((address_space(1))) v4i*)(gptr),                        \
      (__attribute__((address_space(3))) v4i*)(lptr), 0, 0)

__device__ __forceinline__ void wait_asynccnt0() {
  asm volatile("s_wait_asynccnt 0x0" ::: "memory");
}

// ---------------------------------------------------------------------------
// Zero a float buffer (for BN stat accumulators)
// ---------------------------------------------------------------------------
__global__ void zero_kernel(float* __restrict__ p, int n) {
  for (int i = threadIdx.x; i < n; i += blockDim.x) p[i] = 0.0f;
}

// ---------------------------------------------------------------------------
// 3-NN: one thread per query point; xyz2 of the batch staged in LDS.
// ---------------------------------------------------------------------------
__global__ __launch_bounds__(256) void knn_kernel(const float* __restrict__ xyz1,
                                                  const float* __restrict__ xyz2,
                                                  int* __restrict__ idx_out,
                                                  float* __restrict__ w_out) {
  __shared__ float s2[S_ * 3];
  const int chunksPerBatch = N_ / 256;                 // 16
  const int b = blockIdx.x / chunksPerBatch;
  const int chunk = blockIdx.x % chunksPerBatch;

  for (int j = threadIdx.x; j < S_ * 3; j += 256)
    s2[j] = xyz2[(size_t)b * S_ * 3 + j];
  __syncthreads();

  const int n = chunk * 256 + threadIdx.x;
  const size_t row = (size_t)b * N_ + n;
  const float qx = xyz1[row * 3 + 0];
  const float qy = xyz1[row * 3 + 1];
  const float qz = xyz1[row * 3 + 2];

  float d0 = 3.4e38f, d1 = 3.4e38f, d2 = 3.4e38f;
  int i0 = 0, i1 = 0, i2 = 0;
  for (int s = 0; s < S_; ++s) {
    const float dx = qx - s2[s * 3 + 0];
    const float dy = qy - s2[s * 3 + 1];
    const float dz = qz - s2[s * 3 + 2];
    const float d = dx * dx + dy * dy + dz * dz;
    if (d < d0)      { d2 = d1; i2 = i1; d1 = d0; i1 = i0; d0 = d; i0 = s; }
    else if (d < d1) { d2 = d1; i2 = i1; d1 = d;  i1 = s; }
    else if (d < d2) { d2 = d;  i2 = s; }
  }
  float w0 = 1.0f / (d0 + 1e-8f);
  float w1 = 1.0f / (d1 + 1e-8f);
  float w2 = 1.0f / (d2 + 1e-8f);
  const float inv = 1.0f / (w0 + w1 + w2);
  idx_out[row * 3 + 0] = i0;
  idx_out[row * 3 + 1] = i1;
  idx_out[row * 3 + 2] = i2;
  w_out[row * 3 + 0] = w0 * inv;
  w_out[row * 3 + 1] = w1 * inv;
  w_out[row * 3 + 2] = w2 * inv;
}

// ---------------------------------------------------------------------------
// Build Xcat[row][0:384] = concat(points1[b,n,:], interp[b,n,:])
// ---------------------------------------------------------------------------
__global__ __launch_bounds__(128) void interp_kernel(const float* __restrict__ points1,
                                                     const float* __restrict__ points2,
                                                     const int* __restrict__ idx,
                                                     const float* __restrict__ w,
                                                     float* __restrict__ Xcat) {
  const size_t row = blockIdx.x;
  const int b = (int)(row >> 12);        // N_ = 4096
  const int n = (int)(row & (N_ - 1));

  const float* __restrict__ p1 = points1 + ((size_t)b * N_ + n) * D1_;
  float* __restrict__ xr = Xcat + row * CIN;
  xr[threadIdx.x] = p1[threadIdx.x];     // 128 threads cover D1_

  const int j0 = idx[row * 3 + 0];
  const int j1 = idx[row * 3 + 1];
  const int j2 = idx[row * 3 + 2];
  const float w0 = w[row * 3 + 0];
  const float w1 = w[row * 3 + 1];
  const float w2 = w[row * 3 + 2];
  const float* __restrict__ base = points2 + (size_t)b * S_ * D2_;
  const float* __restrict__ r0 = base + (size_t)j0 * D2_;
  const float* __restrict__ r1 = base + (size_t)j1 * D2_;
  const float* __restrict__ r2 = base + (size_t)j2 * D2_;
#pragma unroll
  for (int c = threadIdx.x; c < D2_; c += 128)
    xr[D1_ + c] = w0 * r0[c] + w1 * r1[c] + w2 * r2[c];
}

// ---------------------------------------------------------------------------
// WMMA fp32 GEMM, double-buffered with async global->LDS staging.
//   Y[m][n] = sum_k A'[m][k] * Wt[n][k] + bias[n]
//   A' = A               (BNA == false)  -> A staged via GLOBAL_LOAD_ASYNC_TO_LDS
//   A' = relu(A*s + t)   (BNA == true)   -> A staged via regs (fused BN1), B async
// Block tile 128x64, 8 waves, each wave 32x32 via 2x2 V_WMMA_F32_16X16X4_F32.
// ---------------------------------------------------------------------------
template <int KDIM, int NDIM, bool BNA>
__global__ __launch_bounds__(256) void gemm_kernel(const float* __restrict__ A,
                                                   const float* __restrict__ Wt,
                                                   const float* __restrict__ bias,
                                                   const float* __restrict__ aScale,
                                                   const float* __restrict__ aShift,
                                                   float* __restrict__ Y) {
  constexpr int BM = 128, BN = 64, BK = 32;
  constexpr int LDA = BK + 2, LDB = BK + 2;   // pad: conflict-free b64 fragment reads
  constexpr int NK = KDIM / BK;
  __shared__ __align__(16) float As[2][BM * LDA];
  __shared__ __align__(16) float Bs[2][BN * LDB];

  const int tid  = threadIdx.x;
  const int lane = tid & 31;
  const int wave = tid >> 5;
  const int wm = wave & 3;           // 4 waves along M
  const int wn = wave >> 2;          // 2 waves along N
  const int h   = lane >> 4;         // lane half selects K pair / M+8
  const int l15 = lane & 15;
  const int blockN = blockIdx.x * BN;
  const int blockM = blockIdx.y * BM;

  // ---- staging helpers -----------------------------------------------------
  auto stageB_async = [&](int buf, int k0) {
#pragma unroll
    for (int i = 0; i < 2; ++i) {          // 512 float4 tile, 2 per thread
      const int q = tid + i * 256;
      const int r = q >> 3, c4 = (q & 7) * 4;
      ASYNC_LOAD_B128(Wt + (size_t)(blockN + r) * KDIM + k0 + c4,
                      &Bs[buf][r * LDB + c4]);
    }
  };
  auto stageA_async = [&](int buf, int k0) {
#pragma unroll
    for (int i = 0; i < 4; ++i) {          // 1024 float4 tile, 4 per thread
      const int q = tid + i * 256;
      const int r = q >> 3, c4 = (q & 7) * 4;
      ASYNC_LOAD_B128(A + (size_t)(blockM + r) * KDIM + k0 + c4,
                      &As[buf][r * LDA + c4]);
    }
  };
  auto fetchA_regs = [&](float4* rg, float4* sc, float4* sh, int k0) {
#pragma unroll
    for (int i = 0; i < 4; ++i) {
      const int q = tid + i * 256;
      const int r = q >> 3, c4 = (q & 7) * 4;
      rg[i] = *(const float4*)(A + (size_t)(blockM + r) * KDIM + k0 + c4);
      sc[i] = *(const float4*)(aScale + k0 + c4);
      sh[i] = *(const float4*)(aShift + k0 + c4);
    }
  };
  auto commitA_regs = [&](int buf, const float4* rg, const float4* sc, const float4* sh) {
#pragma unroll
    for (int i = 0; i < 4; ++i) {
      const int q = tid + i * 256;
      const int r = q >> 3, c4 = (q & 7) * 4;
      float4 v = rg[i];
      v.x = fmaxf(v.x * sc[i].x + sh[i].x, 0.0f);
      v.y = fmaxf(v.y * sc[i].y + sh[i].y, 0.0f);
      v.z = fmaxf(v.z * sc[i].z + sh[i].z, 0.0f);
      v.w = fmaxf(v.w * sc[i].w + sh[i].w, 0.0f);
      float* d = &As[buf][r * LDA + c4];
      d[0] = v.x; d[1] = v.y; d[2] = v.z; d[3] = v.w;
    }
  };

  v8f acc[2][2];
#pragma unroll
  for (int i = 0; i < 2; ++i)
#pragma unroll
    for (int j = 0; j < 2; ++j) acc[i][j] = {};

  auto compute = [&](int buf) {
#pragma unroll
    for (int kk = 0; kk < BK; kk += 4) {
      const v2f a0 = *(const v2f*)&As[buf][(wm * 32 +  0 + l15) * LDA + kk + 2 * h];
      const v2f a1 = *(const v2f*)&As[buf][(wm * 32 + 16 + l15) * LDA + kk + 2 * h];
      const v2f b0 = *(const v2f*)&Bs[buf][(wn * 32 +  0 + l15) * LDB + kk + 2 * h];
      const v2f b1 = *(const v2f*)&Bs[buf][(wn * 32 + 16 + l15) * LDB + kk + 2 * h];
      acc[0][0] = __builtin_amdgcn_wmma_f32_16x16x4_f32(false, a0, false, b0, (short)0, acc[0][0], false, false);
      acc[0][1] = __builtin_amdgcn_wmma_f32_16x16x4_f32(false, a0, false, b1, (short)0, acc[0][1], false, false);
      acc[1][0] = __builtin_amdgcn_wmma_f32_16x16x4_f32(false, a1, false, b0, (short)0, acc[1][0], false, false);
      acc[1][1] = __builtin_amdgcn_wmma_f32_16x16x4_f32(false, a1, false, b1, (short)0, acc[1][1], false, false);
    }
  };

  // ---- prologue: stage chunk 0 into buffer 0 -------------------------------
  float4 rg[4], sc[4], sh[4];
  stageB_async(0, 0);
  if constexpr (!BNA) {
    stageA_async(0, 0);
  } else {
    fetchA_regs(rg, sc, sh, 0);
    commitA_regs(0, rg, sc, sh);
  }
  wait_asynccnt0();
  __syncthreads();

  // ---- pipelined main loop -------------------------------------------------
  for (int kc = 0; kc < NK; ++kc) {
    const int cur = kc & 1, nxt = cur ^ 1;
    const int k1 = (kc + 1) * BK;
    const bool more = (kc + 1) < NK;
    if (more) {
      stageB_async(nxt, k1);                 // overlap with compute below
      if constexpr (!BNA) stageA_async(nxt, k1);
      else                fetchA_regs(rg, sc, sh, k1);
    }
    compute(cur);
    if (BNA && more) commitA_regs(nxt, rg, sc, sh);
    wait_asynccnt0();                        // own wave's async copies landed
    __syncthreads();                         // everyone's tile visible
  }

  // ---- epilogue: D layout VGPR r -> m = r + 8h, n = l15 --------------------
#pragma unroll
  for (int i = 0; i < 2; ++i)
#pragma unroll
    for (int j = 0; j < 2; ++j) {
      const int col = blockN + wn * 32 + j * 16 + l15;
      const float bv = bias[col];
#pragma unroll
      for (int r = 0; r < 8; ++r) {
        const int rowg = blockM + wm * 32 + i * 16 + r + 8 * h;
        Y[(size_t)rowg * NDIM + col] = acc[i][j][r] + bv;
      }
    }
}

// ---------------------------------------------------------------------------
// Per-channel sum / sum-of-squares over all rows (atomic partials per block)
// ---------------------------------------------------------------------------
template <int C>
__global__ void colstats_kernel(const float* __restrict__ Y,
                                float* __restrict__ sum, float* __restrict__ sumsq) {
  constexpr int ROWS = 256;
  const int c = threadIdx.x;
  const size_t r0 = (size_t)blockIdx.x * ROWS;
  float s = 0.0f, sq = 0.0f;
  for (int r = 0; r < ROWS; ++r) {
    const float v = Y[(r0 + r) * C + c];
    s += v; sq += v * v;
  }
  atomicAdd(&sum[c], s);
  atomicAdd(&sumsq[c], sq);
}

// ---------------------------------------------------------------------------
// Fold BN (training-mode batch stats, biased var) + gamma/beta into scale/shift
// ---------------------------------------------------------------------------
__global__ void bn_finalize_kernel(const float* __restrict__ sum, const float* __restrict__ sumsq,
                                   const float* __restrict__ g, const float* __restrict__ be,
                                   float* __restrict__ scale, float* __restrict__ shift, int C) {
  const int c = threadIdx.x;
  if (c < C) {
    const float invM = 1.0f / (float)M_TOT;
    const float mean = sum[c] * invM;
    const float var  = sumsq[c] * invM - mean * mean;
    const float scv  = g[c] * rsqrtf(var + BN_EPS);
    scale[c] = scv;
    shift[c] = be[c] - mean * scv;
  }
}

// ---------------------------------------------------------------------------
// Final BN2 + ReLU -> output [B, N, COUT] row-major
// ---------------------------------------------------------------------------
__global__ __launch_bounds__(256) void bn_relu_out_kernel(const float* __restrict__ Y2,
                                                          const float* __restrict__ scale,
                                                          const float* __restrict__ shift,
                                                          float* __restrict__ out) {
  const size_t i = (size_t)blockIdx.x * blockDim.x + threadIdx.x;
  const int c = (int)(i & (COUT - 1));
  out[i] = fmaxf(Y2[i] * scale[c] + shift[c], 0.0f);
}

// ---------------------------------------------------------------------------
extern "C" void kernel_launch(void* const* d_in, const int* in_sizes, int n_in,
                              void* d_out, int out_size, void* d_ws, size_t ws_size,
                              hipStream_t stream) {
  (void)in_sizes; (void)n_in; (void)out_size; (void)ws_size;
  const float* xyz1    = (const float*)d_in[0];
  const float* xyz2    = (const float*)d_in[1];
  const float* points1 = (const float*)d_in[2];
  const float* points2 = (const float*)d_in[3];
  const float* W1      = (const float*)d_in[4];   // [256, 384]
  const float* b1      = (const float*)d_in[5];
  const float* g1      = (const float*)d_in[6];
  const float* be1     = (const float*)d_in[7];
  const float* W2      = (const float*)d_in[8];   // [128, 256]
  const float* b2      = (const float*)d_in[9];
  const float* g2      = (const float*)d_in[10];
  const float* be2     = (const float*)d_in[11];
  float* out = (float*)d_out;

  // workspace carve-up (16B alignment everywhere)
  char* p = (char*)d_ws;
  int*   idx_ws = (int*)p;    p += (size_t)M_TOT * 3 * sizeof(int);
  float* w_ws   = (float*)p;  p += (size_t)M_TOT * 3 * sizeof(float);
  float* Xcat   = (float*)p;  p += (size_t)M_TOT * CIN  * sizeof(float);
  float* Y1     = (float*)p;  p += (size_t)M_TOT * CMID * sizeof(float);
  float* Y2     = (float*)p;  p += (size_t)M_TOT * COUT * sizeof(float);
  float* stats  = (float*)p;                       // contiguous accumulators
  float* sum1 = stats;        float* sq1 = stats + 256;
  float* sum2 = stats + 512;  float* sq2 = stats + 640;   // 768 floats to zero
  float* scale1 = stats + 768;  float* shift1 = stats + 1024;
  float* scale2 = stats + 1280; float* shift2 = stats + 1408;

  zero_kernel<<<1, 256, 0, stream>>>(stats, 768);

  knn_kernel<<<dim3(B_ * (N_ / 256)), dim3(256), 0, stream>>>(xyz1, xyz2, idx_ws, w_ws);

  interp_kernel<<<dim3(M_TOT), dim3(128), 0, stream>>>(points1, points2, idx_ws, w_ws, Xcat);

  gemm_kernel<CIN, CMID, false>
      <<<dim3(CMID / 64, M_TOT / 128), dim3(256), 0, stream>>>(Xcat, W1, b1, nullptr, nullptr, Y1);

  colstats_kernel<CMID><<<dim3(M_TOT / 256), dim3(CMID), 0, stream>>>(Y1, sum1, sq1);
  bn_finalize_kernel<<<1, 256, 0, stream>>>(sum1, sq1, g1, be1, scale1, shift1, CMID);

  gemm_kernel<CMID, COUT, true>
      <<<dim3(COUT / 64, M_TOT / 128), dim3(256), 0, stream>>>(Y1, W2, b2, scale1, shift1, Y2);

  colstats_kernel<COUT><<<dim3(M_TOT / 256), dim3(COUT), 0, stream>>>(Y2, sum2, sq2);
  bn_finalize_kernel<<<1, 256, 0, stream>>>(sum2, sq2, g2, be2, scale2, shift2, COUT);

  bn_relu_out_kernel<<<dim3((M_TOT * COUT) / 256), dim3(256), 0, stream>>>(Y2, scale2, shift2, out);
}